// LightGCN_27195732918361
// MI455X (gfx1250) — compile-verified
//
#include <hip/hip_runtime.h>
#include <hip/hip_bf16.h>
#include <stdint.h>

#define N_USERS   100000
#define M_ITEMS   50000
#define N_NODES   (N_USERS + M_ITEMS)
#define DIM       64
#define NODE_FLOATS ((size_t)N_NODES * DIM)

// ---- TDM descriptor vector types (amdgpu-toolchain clang-23: 6-arg builtin) ----
typedef __attribute__((ext_vector_type(4))) unsigned int v4u;
typedef __attribute__((ext_vector_type(8))) int          v8i;
typedef __attribute__((ext_vector_type(4))) int          v4i;

// Native fp32 global atomic add (no return) -> global_atomic_add_f32
__device__ __forceinline__ void atomic_add_f32(float* p, float v) {
  asm volatile("global_atomic_add_f32 %0, %1, off" :: "v"(p), "v"(v) : "memory");
}

// ---------------- init: acc = h = concat(user_emb, item_emb) ----------------
__global__ void k_init(const float* __restrict__ ue, const float* __restrict__ ie,
                       float* __restrict__ acc, float* __restrict__ h, size_t n4) {
  size_t i = (size_t)blockIdx.x * blockDim.x + threadIdx.x;
  if (i >= n4) return;
  size_t u4 = (size_t)N_USERS * DIM / 4;
  float4 v = (i < u4) ? ((const float4*)ue)[i] : ((const float4*)ie)[i - u4];
  ((float4*)acc)[i] = v;
  ((float4*)h)[i]   = v;
}

__global__ void k_zero(float* __restrict__ p, size_t n4) {
  size_t i = (size_t)blockIdx.x * blockDim.x + threadIdx.x;
  if (i >= n4) return;
  ((float4*)p)[i] = make_float4(0.f, 0.f, 0.f, 0.f);
}

// acc = (acc + h) * s   (s = 1 for inner layers, 0.25 folded into last layer)
__global__ void k_add_scale(float* __restrict__ acc, const float* __restrict__ h,
                            float s, size_t n4) {
  size_t i = (size_t)blockIdx.x * blockDim.x + threadIdx.x;
  if (i >= n4) return;
  float4 a = ((const float4*)acc)[i];
  float4 b = ((const float4*)h)[i];
  a.x = (a.x + b.x) * s; a.y = (a.y + b.y) * s;
  a.z = (a.z + b.z) * s; a.w = (a.w + b.w) * s;
  ((float4*)acc)[i] = a;
}

// ---------------- SpMM with TDM row-gather ----------------
#define SPMM_BLOCK      256
#define WAVES_PER_BLOCK 8
#define EDGES_PER_WAVE  64
#define EDGES_PER_BLOCK (WAVES_PER_BLOCK * EDGES_PER_WAVE)   // 512
#define GROUP_ROWS      8
#define ROW_F           64
#define BUF_FLOATS      (GROUP_ROWS * ROW_F)                 // 512 floats = 2KB

// Issue one TDM gather of up to 8 rows of x (64 fp32 each) into LDS.
__device__ __forceinline__ void tdm_gather8(uint64_t xbase, unsigned lds_byte,
                                            const int* __restrict__ esrc,
                                            int eg, int nvalid) {
  int idx[8];
  if (nvalid == GROUP_ROWS) {
    int4 ia = *(const int4*)(esrc + eg);
    int4 ib = *(const int4*)(esrc + eg + 4);
    idx[0] = ia.x; idx[1] = ia.y; idx[2] = ia.z; idx[3] = ia.w;
    idx[4] = ib.x; idx[5] = ib.y; idx[6] = ib.z; idx[7] = ib.w;
  } else {
#pragma unroll
    for (int j = 0; j < 8; ++j) idx[j] = (j < nvalid) ? esrc[eg + j] : 0;
  }
#pragma unroll
  for (int j = 0; j < 8; ++j) idx[j] = __builtin_amdgcn_readfirstlane(idx[j]);

  // D# group 0: count=1 | gather_mode | 32-bit indices; lds addr; 57b global addr; type=2
  v4u g0;
  g0[0] = 0xC0000001u;
  g0[1] = lds_byte;
  g0[2] = (unsigned)xbase;
  g0[3] = ((unsigned)(xbase >> 32) & 0x01FFFFFFu) | 0x80000000u;

  // D# group 1
  v8i g1;
  g1[0] = 0x00020000;                                   // data_size=4B, no mask/flags
  g1[1] = (int)((unsigned)ROW_F << 16);                 // tensor_dim0 = 64 (lo16)
  g1[2] = (int)(((unsigned)N_NODES & 0xFFFFu) << 16);   // tensor_dim1 lo16
  g1[3] = (int)(((unsigned)N_NODES >> 16) | ((unsigned)ROW_F << 16)); // dim1 hi16 | tile_dim0
  g1[4] = nvalid;                                       // tile_dim1 = #valid indices
  g1[5] = ROW_F;                                        // tensor_dim0_stride = 64
  g1[6] = 0;
  g1[7] = 0;

  v4i g2; g2[0] = idx[0]; g2[1] = idx[1]; g2[2] = idx[2]; g2[3] = idx[3];
  v4i g3; g3[0] = idx[4]; g3[1] = idx[5]; g3[2] = idx[6]; g3[3] = idx[7];

  v8i g4 = (v8i)(0);   // extra group required by the 6-arg builtin (zero-filled)

  __builtin_amdgcn_tensor_load_to_lds(g0, g1, g2, g3, g4, 0);
}

__global__ __launch_bounds__(SPMM_BLOCK) void k_spmm(
    const float* __restrict__ x,
    const int*   __restrict__ esrc,
    const int*   __restrict__ edst,
    const float* __restrict__ eval,
    float*       __restrict__ y,
    int nEdges)
{
  __shared__ float smem[WAVES_PER_BLOCK * 2 * BUF_FLOATS];   // 32 KB

  int lane = threadIdx.x & 31;
  int wave = __builtin_amdgcn_readfirstlane((int)(threadIdx.x >> 5));
  int e0 = blockIdx.x * EDGES_PER_BLOCK + wave * EDGES_PER_WAVE;
  if (e0 >= nEdges) return;
  int rem = nEdges - e0; if (rem > EDGES_PER_WAVE) rem = EDGES_PER_WAVE;
  int ngroups = (rem + GROUP_ROWS - 1) >> 3;

  float* wbase = &smem[wave * 2 * BUF_FLOATS];
  unsigned lds0 = (unsigned)(uint64_t)(void*)wbase;  // low 32 bits = LDS byte offset
  const unsigned bufstride = BUF_FLOATS * 4;         // 2048 B
  uint64_t xa = (uint64_t)(const void*)x;

  { int nv0 = rem > GROUP_ROWS ? GROUP_ROWS : rem;
    tdm_gather8(xa, lds0, esrc, e0, nv0); }

  for (int g = 0; g < ngroups; ++g) {
    int eg = e0 + g * GROUP_ROWS;
    bool more = (g + 1 < ngroups);
    if (more) {
      int nvn = rem - (g + 1) * GROUP_ROWS;
      if (nvn > GROUP_ROWS) nvn = GROUP_ROWS;
      tdm_gather8(xa, lds0 + ((unsigned)((g + 1) & 1)) * bufstride,
                  esrc, e0 + (g + 1) * GROUP_ROWS, nvn);
      __builtin_amdgcn_s_wait_tensorcnt(1);   // previous tile landed, next in flight
    } else {
      __builtin_amdgcn_s_wait_tensorcnt(0);
    }

    const float* buf = wbase + (g & 1) * BUF_FLOATS;
    int nv = rem - g * GROUP_ROWS; if (nv > GROUP_ROWS) nv = GROUP_ROWS;

#pragma unroll
    for (int it = 0; it < 4; ++it) {
      int unit = it * 32 + lane;     // 0..127 : 8 rows x 16 float4 chunks
      int r = unit >> 4;
      int c = unit & 15;
      if (r < nv) {
        int e = eg + r;
        float vv = __builtin_nontemporal_load(eval + e);   // streamed once -> NT
        int  dst = __builtin_nontemporal_load(edst + e);
        float4 xv = *(const float4*)(buf + r * ROW_F + c * 4);
        float* op = y + (size_t)dst * ROW_F + c * 4;
        atomic_add_f32(op + 0, xv.x * vv);
        atomic_add_f32(op + 1, xv.y * vv);
        atomic_add_f32(op + 2, xv.z * vv);
        atomic_add_f32(op + 3, xv.w * vv);
      }
    }
    if (lane == 0) __builtin_prefetch(esrc + eg + 4096, 0, 1);  // edge-stream look-ahead
  }
}

// ---------------- driver ----------------
extern "C" void kernel_launch(void* const* d_in, const int* in_sizes, int n_in,
                              void* d_out, int out_size, void* d_ws, size_t ws_size,
                              hipStream_t stream) {
  const float* ue   = (const float*)d_in[0];
  const float* ie   = (const float*)d_in[1];
  const int*   esrc = (const int*)d_in[2];
  const int*   edst = (const int*)d_in[3];
  const float* eval = (const float*)d_in[4];
  int nE = in_sizes[2];

  float* out = (float*)d_out;
  float* h0  = (float*)d_ws;
  float* h1  = h0 + NODE_FLOATS;

  size_t n4 = NODE_FLOATS / 4;
  const int TB = 256;
  int nb_node = (int)((n4 + TB - 1) / TB);

  k_init<<<nb_node, TB, 0, stream>>>(ue, ie, out, h0, n4);

  float* hcur = h0; float* hnext = h1;
  for (int layer = 0; layer < 3; ++layer) {
    k_zero<<<nb_node, TB, 0, stream>>>(hnext, n4);
    int nb_e = (nE + EDGES_PER_BLOCK - 1) / EDGES_PER_BLOCK;
    k_spmm<<<nb_e, SPMM_BLOCK, 0, stream>>>(hcur, esrc, edst, eval, hnext, nE);
    float s = (layer == 2) ? 0.25f : 1.0f;
    k_add_scale<<<nb_node, TB, 0, stream>>>(out, hnext, s, n4);
    float* t = hcur; hcur = hnext; hnext = t;
  }
}